// CTCLayer_9612136808817
// MI455X (gfx1250) — compile-verified
//
#include <hip/hip_runtime.h>
#include <stdint.h>

// CTC forward loss, B=1024 T=256 C=128 L=32 (S=65 states).
// Bandwidth-bound: 134 MB of y_pred read once -> ~5.8us floor @ 23.3 TB/s.
// One wave32 per batch; lane l owns states {2l, 2l+1}; state 64 carried
// redundantly. Rows are software-pipelined into LDS with the CDNA5 async
// copy engine (global_load_async_to_lds_b128 / s_wait_asynccnt), 8 deep,
// non-temporal, so the serially-dependent T-recursion never stalls on HBM.

#define TT  256
#define CC  128
#define LL  32
#define NBUF 8                                       // async pipeline depth
#define NEGF (-1e30f)
#define EPSF (1e-7f)

__device__ __forceinline__ float lae2(float a, float b) {
  float m = fmaxf(a, b);
  return m + __logf(__expf(a - m) + __expf(b - m));   // one exp is exactly 1
}
__device__ __forceinline__ float lae3(float a, float b, float c) {
  float m = fmaxf(fmaxf(a, b), c);
  return m + __logf(__expf(a - m) + __expf(b - m) + __expf(c - m));
}

__global__ __launch_bounds__(32)
void ctc_loss_fwd(const int* __restrict__ y_true,
                  const float* __restrict__ y_pred,
                  float* __restrict__ out) {
  __shared__ float buf[NBUF][CC];                    // 8-deep row pipeline, 4KB
  const int b    = blockIdx.x;
  const int lane = threadIdx.x;

  const float* base = y_pred + (size_t)b * TT * CC;
  // generic pointer to __shared__: low 32 bits are the LDS byte offset
  const uint32_t lds_base = (uint32_t)(uintptr_t)(&buf[0][0]);

  // each lane async-copies 16B of the 512B row: slot = LDS buffer index.
  // NT hint: y_pred is a read-once 134MB stream, keep it out of L2.
  auto issue = [&](int row, int slot) {
    uint64_t g = (uint64_t)(uintptr_t)(base + (size_t)row * CC + lane * 4);
    uint32_t l = lds_base + (uint32_t)((slot * CC + lane * 4) * 4);
    asm volatile("global_load_async_to_lds_b128 %0, %1, off th:TH_LOAD_NT"
                 :: "v"(l), "v"(g) : "memory");
  };

  #pragma unroll
  for (int p = 0; p < NBUF - 1; ++p) issue(p, p);    // prime the pipeline

  // extended sequence: ext[2l]=blank(=C-1), ext[2l+1]=label[l]; ext[64]=blank
  const int lab  = y_true[(size_t)b * LL + lane];
  const int labp = __shfl_up(lab, 1);
  const bool skip = (lane > 0) && (lab != labp);     // skip_ok for odd states

  float a0 = NEGF, a1 = NEGF, a64 = NEGF;

  for (int t = 0; t < TT; ++t) {
    int rn = t + NBUF - 1;
    rn = (rn < TT) ? rn : (TT - 1);                  // clamped tail re-loads
    issue(rn, (t + NBUF - 1) & (NBUF - 1));
    asm volatile("s_wait_asynccnt 7" ::: "memory");  // row t has landed in LDS

    const float* row = &buf[t & (NBUF - 1)][0];
    const float lpb = __logf(row[CC - 1] + EPSF);    // blank logp (broadcast)
    const float lp1 = __logf(row[lab] + EPSF);       // this lane's label logp

    if (t == 0) {
      a0  = (lane == 0) ? lpb : NEGF;                // alpha0[0]
      a1  = (lane == 0) ? lp1 : NEGF;                // alpha0[1]
      a64 = NEGF;
    } else {
      const float a1_31 = __shfl(a1, 31);            // old alpha[63]
      float am1 = __shfl_up(a1, 1);                  // old alpha[2l-1]
      if (lane == 0) am1 = NEGF;
      const float na0  = lae2(a0, am1) + lpb;                         // s=2l
      const float na1  = lae3(a1, a0, skip ? am1 : NEGF) + lp1;       // s=2l+1
      const float na64 = lae2(a64, a1_31) + lpb;                      // s=64
      a0 = na0; a1 = na1; a64 = na64;
    }
  }

  const float a63 = __shfl(a1, 31);
  if (lane == 0) out[b] = -lae2(a64, a63);
}

extern "C" void kernel_launch(void* const* d_in, const int* in_sizes, int n_in,
                              void* d_out, int out_size, void* d_ws, size_t ws_size,
                              hipStream_t stream) {
  (void)n_in; (void)d_ws; (void)ws_size; (void)out_size;
  const int* y_true   = (const int*)d_in[0];         // [B, L] labels
  const float* y_pred = (const float*)d_in[1];       // [B, T, C] probabilities
  float* out = (float*)d_out;                        // [B, 1] loss
  const int B = in_sizes[0] / LL;
  ctc_loss_fwd<<<dim3(B), dim3(32), 0, stream>>>(y_true, y_pred, out);
}